// GruActionEncoder_81200651698475
// MI455X (gfx1250) — compile-verified
//
#include <hip/hip_runtime.h>
#include <math.h>

// Problem constants (from reference): N=256, T=128, H=1024, A=16
#define NB 256
#define TT 128
#define HH 1024
#define AA 16

typedef __attribute__((ext_vector_type(2))) float v2f;
typedef __attribute__((ext_vector_type(8))) float v8f;

__device__ __forceinline__ v8f wmma4(v2f a, v2f b, v8f c) {
  // D = A(16x4,f32) * B(4x16,f32) + C(16x16,f32)
  return __builtin_amdgcn_wmma_f32_16x16x4_f32(
      /*neg_a=*/false, a, /*neg_b=*/false, b,
      /*c_mod=*/(short)0, c, /*reuse_a=*/false, /*reuse_b=*/false);
}

__device__ __forceinline__ v8f splat8(float x) {
  v8f r;
#pragma unroll
  for (int i = 0; i < 8; ++i) r[i] = x;
  return r;
}

__device__ __forceinline__ float sigmoidf_(float x) {
  return 1.0f / (1.0f + __expf(-x));
}

// ---------------------------------------------------------------------------
// Kernel: zero-init h state
// ---------------------------------------------------------------------------
__global__ void gru_zero_kernel(float* __restrict__ p, int n) {
  int i = blockIdx.x * blockDim.x + threadIdx.x;
  if (i < n) p[i] = 0.0f;
}

// ---------------------------------------------------------------------------
// Kernel A: Gi[t_local, n, 0:3H] = emb[n, t0+t_local, :] @ W_ih^T + b_ih
// One wave -> 32 x 64 output macro-tile (2 row-tiles x 4 col-tiles,
// 8 accumulators). Per k-step: 2 A-frag + 4 B-frag loads, 8 WMMAs.
// Grid = tc*48 blocks of 8 waves (tc*384 macro-tiles).
// ---------------------------------------------------------------------------
__global__ void gru_gi_gemm(const float* __restrict__ emb,
                            const float* __restrict__ Wih,
                            const float* __restrict__ bih,
                            float* __restrict__ Gi,
                            int t0) {
  const int lane = threadIdx.x & 31;
  const int wave = threadIdx.x >> 5;
  const int strip = blockIdx.x * 8 + wave;
  const int ncolstrip = (3 * HH) / 64;  // 48
  const int rowpair = strip / ncolstrip;
  const int colstrip = strip % ncolstrip;
  const int row0 = rowpair * 32;   // chunk row (t_local*256 + n); 32 | 256 so no t straddle
  const int col0 = colstrip * 64;

  const int m = lane & 15;
  const int khalf = (lane >> 4) * 2;
  const int ncol = lane & 15;

  const int t = t0 + row0 / NB;
  const int n0 = (row0 % NB) + m;
  const float* arow0 = emb + ((size_t)n0 * TT + t) * HH + khalf;
  const float* arow1 = emb + ((size_t)(n0 + 16) * TT + t) * HH + khalf;

  const float* b0 = Wih + (size_t)(col0 + 0 * 16 + ncol) * HH + khalf;
  const float* b1 = Wih + (size_t)(col0 + 1 * 16 + ncol) * HH + khalf;
  const float* b2 = Wih + (size_t)(col0 + 2 * 16 + ncol) * HH + khalf;
  const float* b3 = Wih + (size_t)(col0 + 3 * 16 + ncol) * HH + khalf;

  const float bv0 = bih[col0 + 0 * 16 + ncol];
  const float bv1 = bih[col0 + 1 * 16 + ncol];
  const float bv2 = bih[col0 + 2 * 16 + ncol];
  const float bv3 = bih[col0 + 3 * 16 + ncol];
  v8f p0 = splat8(bv0), p1 = splat8(bv1), p2 = splat8(bv2), p3 = splat8(bv3);
  v8f q0 = splat8(bv0), q1 = splat8(bv1), q2 = splat8(bv2), q3 = splat8(bv3);

  for (int k = 0; k < HH; k += 4) {
    v2f a0 = *(const v2f*)(arow0 + k);
    v2f a1 = *(const v2f*)(arow1 + k);
    v2f w0 = *(const v2f*)(b0 + k);
    v2f w1 = *(const v2f*)(b1 + k);
    v2f w2 = *(const v2f*)(b2 + k);
    v2f w3 = *(const v2f*)(b3 + k);
    p0 = wmma4(a0, w0, p0);
    p1 = wmma4(a0, w1, p1);
    p2 = wmma4(a0, w2, p2);
    p3 = wmma4(a0, w3, p3);
    q0 = wmma4(a1, w0, q0);
    q1 = wmma4(a1, w1, q1);
    q2 = wmma4(a1, w2, q2);
    q3 = wmma4(a1, w3, q3);
  }

  const int srow0 = row0 + (lane >> 4) * 8;
#pragma unroll
  for (int v = 0; v < 8; ++v) {
    size_t r = (size_t)(srow0 + v) * (3 * HH) + col0 + ncol;
    Gi[r + 0 * 16] = p0[v];
    Gi[r + 1 * 16] = p1[v];
    Gi[r + 2 * 16] = p2[v];
    Gi[r + 3 * 16] = p3[v];
    size_t r2 = r + (size_t)16 * (3 * HH);
    Gi[r2 + 0 * 16] = q0[v];
    Gi[r2 + 1 * 16] = q1[v];
    Gi[r2 + 2 * 16] = q2[v];
    Gi[r2 + 3 * 16] = q3[v];
  }
}

// ---------------------------------------------------------------------------
// Kernel B: henc = [h | a_t] @ W_enc^T + b_enc     (256 x 1024, K = 1040)
// One wave -> 16 x 64 strip (kept narrow: maximize wave count in the
// serial chain). Grid = 32 blocks of 8 waves (256 strips).
// ---------------------------------------------------------------------------
__global__ void gru_henc_gemm(const float* __restrict__ h,
                              const float* __restrict__ act,
                              const float* __restrict__ Wenc,
                              const float* __restrict__ benc,
                              float* __restrict__ henc,
                              int t) {
  const int lane = threadIdx.x & 31;
  const int wave = threadIdx.x >> 5;
  const int strip = blockIdx.x * 8 + wave;
  const int ncolstrip = HH / 64;  // 16
  const int rowtile = strip / ncolstrip;
  const int colstrip = strip % ncolstrip;
  const int row0 = rowtile * 16;
  const int col0 = colstrip * 64;

  const int m = lane & 15;
  const int khalf = (lane >> 4) * 2;
  const int ncol = lane & 15;
  const int LDB = HH + AA;  // 1040

  const float* hrow = h + (size_t)(row0 + m) * HH + khalf;
  const float* arow = act + ((size_t)(row0 + m) * TT + t) * AA + khalf;

  const float* b0 = Wenc + (size_t)(col0 + 0 * 16 + ncol) * LDB + khalf;
  const float* b1 = Wenc + (size_t)(col0 + 1 * 16 + ncol) * LDB + khalf;
  const float* b2 = Wenc + (size_t)(col0 + 2 * 16 + ncol) * LDB + khalf;
  const float* b3 = Wenc + (size_t)(col0 + 3 * 16 + ncol) * LDB + khalf;

  v8f acc0 = splat8(benc[col0 + 0 * 16 + ncol]);
  v8f acc1 = splat8(benc[col0 + 1 * 16 + ncol]);
  v8f acc2 = splat8(benc[col0 + 2 * 16 + ncol]);
  v8f acc3 = splat8(benc[col0 + 3 * 16 + ncol]);

  for (int k = 0; k < HH; k += 4) {  // h part of the concat
    v2f a = *(const v2f*)(hrow + k);
    acc0 = wmma4(a, *(const v2f*)(b0 + k), acc0);
    acc1 = wmma4(a, *(const v2f*)(b1 + k), acc1);
    acc2 = wmma4(a, *(const v2f*)(b2 + k), acc2);
    acc3 = wmma4(a, *(const v2f*)(b3 + k), acc3);
  }
  for (int k = HH; k < HH + AA; k += 4) {  // action part of the concat
    v2f a = *(const v2f*)(arow + (k - HH));
    acc0 = wmma4(a, *(const v2f*)(b0 + k), acc0);
    acc1 = wmma4(a, *(const v2f*)(b1 + k), acc1);
    acc2 = wmma4(a, *(const v2f*)(b2 + k), acc2);
    acc3 = wmma4(a, *(const v2f*)(b3 + k), acc3);
  }

  const int srow = row0 + (lane >> 4) * 8;
#pragma unroll
  for (int v = 0; v < 8; ++v) {
    size_t r = (size_t)(srow + v) * HH + col0 + ncol;
    henc[r + 0 * 16] = acc0[v];
    henc[r + 1 * 16] = acc1[v];
    henc[r + 2 * 16] = acc2[v];
    henc[r + 3 * 16] = acc3[v];
  }
}

// ---------------------------------------------------------------------------
// Kernel C (fused): gh = henc @ W_hh^T + b_hh for gate cols (c, c+H, c+2H),
// then r/z/n gates + state update, writing h' directly.
// One wave -> 32 rows x 16 h-cols (2 row-tiles x 3 gates = 6 accumulators).
// Per k-step: 2 A-frag + 3 B-frag loads, 6 WMMAs. Grid = 64 blocks (512 waves).
// ---------------------------------------------------------------------------
__global__ void gru_step_fused(const float* __restrict__ henc,
                               const float* __restrict__ Whh,
                               const float* __restrict__ bhh,
                               const float* __restrict__ gi_t,
                               float* __restrict__ hout) {
  const int lane = threadIdx.x & 31;
  const int wave = threadIdx.x >> 5;
  const int strip = blockIdx.x * 8 + wave;
  const int ncoltile = HH / 16;  // 64
  const int rowpair = strip / ncoltile;   // 0..7 (32-row macro tiles)
  const int coltile = strip % ncoltile;
  const int row0 = rowpair * 32;
  const int col0 = coltile * 16;

  const int m = lane & 15;
  const int khalf = (lane >> 4) * 2;
  const int ncol = lane & 15;
  const int col = col0 + ncol;

  const float* arow0 = henc + (size_t)(row0 + m) * HH + khalf;
  const float* arow1 = henc + (size_t)(row0 + 16 + m) * HH + khalf;
  const float* br = Whh + (size_t)(0 * HH + col) * HH + khalf;
  const float* bz = Whh + (size_t)(1 * HH + col) * HH + khalf;
  const float* bn = Whh + (size_t)(2 * HH + col) * HH + khalf;

  const float bvr = bhh[0 * HH + col];
  const float bvz = bhh[1 * HH + col];
  const float bvn = bhh[2 * HH + col];
  v8f pr = splat8(bvr), pz = splat8(bvz), pn = splat8(bvn);
  v8f qr = splat8(bvr), qz = splat8(bvz), qn = splat8(bvn);

  for (int k = 0; k < HH; k += 4) {
    v2f a0 = *(const v2f*)(arow0 + k);
    v2f a1 = *(const v2f*)(arow1 + k);
    v2f wr = *(const v2f*)(br + k);
    v2f wz = *(const v2f*)(bz + k);
    v2f wn = *(const v2f*)(bn + k);
    pr = wmma4(a0, wr, pr);
    pz = wmma4(a0, wz, pz);
    pn = wmma4(a0, wn, pn);
    qr = wmma4(a1, wr, qr);
    qz = wmma4(a1, wz, qz);
    qn = wmma4(a1, wn, qn);
  }

  const int srow = row0 + (lane >> 4) * 8;
#pragma unroll
  for (int v = 0; v < 8; ++v) {
    // row tile 0
    {
      const int row = srow + v;
      const size_t gb = (size_t)row * (3 * HH) + col;
      float i_r = gi_t[gb];
      float i_z = gi_t[gb + HH];
      float i_n = gi_t[gb + 2 * HH];
      float hv = henc[(size_t)row * HH + col];
      float r = sigmoidf_(i_r + pr[v]);
      float z = sigmoidf_(i_z + pz[v]);
      float nn = tanhf(i_n + r * pn[v]);
      hout[(size_t)row * HH + col] = (1.0f - z) * nn + z * hv;
    }
    // row tile 1
    {
      const int row = srow + 16 + v;
      const size_t gb = (size_t)row * (3 * HH) + col;
      float i_r = gi_t[gb];
      float i_z = gi_t[gb + HH];
      float i_n = gi_t[gb + 2 * HH];
      float hv = henc[(size_t)row * HH + col];
      float r = sigmoidf_(i_r + qr[v]);
      float z = sigmoidf_(i_z + qz[v]);
      float nn = tanhf(i_n + r * qn[v]);
      hout[(size_t)row * HH + col] = (1.0f - z) * nn + z * hv;
    }
  }
}

// ---------------------------------------------------------------------------
// Host-side orchestration
// ---------------------------------------------------------------------------
extern "C" void kernel_launch(void* const* d_in, const int* in_sizes, int n_in,
                              void* d_out, int out_size, void* d_ws, size_t ws_size,
                              hipStream_t stream) {
  const float* emb  = (const float*)d_in[0];   // [N, T, H]
  const float* act  = (const float*)d_in[1];   // [N, T, A]
  const float* Wenc = (const float*)d_in[2];   // [H, H+A]
  const float* benc = (const float*)d_in[3];   // [H]
  const float* Wih  = (const float*)d_in[4];   // [3H, H]
  const float* Whh  = (const float*)d_in[5];   // [3H, H]
  const float* bih  = (const float*)d_in[6];   // [3H]
  const float* bhh  = (const float*)d_in[7];   // [3H]
  float* out = (float*)d_out;                  // [N, H]

  // Workspace layout: h state | henc | Gi chunk
  float* h    = (float*)d_ws;
  float* henc = h + (size_t)NB * HH;
  float* Gi   = henc + (size_t)NB * HH;

  const size_t fixedB = 2ull * NB * HH * sizeof(float);        // h + henc
  const size_t perT   = (size_t)NB * 3 * HH * sizeof(float);   // one Gi timestep
  int Tc = 1;
  if (ws_size > fixedB + perT) {
    size_t mx = (ws_size - fixedB) / perT;
    Tc = (int)(mx < (size_t)TT ? mx : (size_t)TT);
  }

  gru_zero_kernel<<<(NB * HH + 255) / 256, 256, 0, stream>>>(h, NB * HH);

  for (int t0 = 0; t0 < TT; t0 += Tc) {
    const int tc = (TT - t0) < Tc ? (TT - t0) : Tc;
    // Parallel precompute of input gates for this chunk (bulk of the FLOPs).
    gru_gi_gemm<<<tc * 48, 256, 0, stream>>>(emb, Wih, bih, Gi, t0);
    // Serial recurrent chain for the chunk.
    for (int tl = 0; tl < tc; ++tl) {
      const int t = t0 + tl;
      gru_henc_gemm<<<32, 256, 0, stream>>>(h, act, Wenc, benc, henc, t);
      float* ho = (t == TT - 1) ? out : h;
      gru_step_fused<<<64, 256, 0, stream>>>(
          henc, Whh, bhh, Gi + (size_t)tl * NB * 3 * HH, ho);
    }
  }
}